// Qwen3NextGatedDeltaNetDecode_28750511079923
// MI455X (gfx1250) — compile-verified
//
#include <hip/hip_runtime.h>
#include <math.h>

#define HK 16
#define HV 32
#define DD 128
#define KK 4
#define CC (DD * (2 * HK + HV)) /* 8192 */

typedef __attribute__((ext_vector_type(2))) float v2f;
typedef __attribute__((ext_vector_type(8))) float v8f;

__device__ __forceinline__ float silu_(float x) { return x / (1.0f + __expf(-x)); }
__device__ __forceinline__ float softplus_(float x) {
    return (x > 20.0f) ? x : log1pf(__expf(x));
}

// One block (128 threads = 4 waves) per (b, h). Streams the 64 KB state tile
// once through v_wmma_f32_16x16x4_f32, computing S.q and S.k simultaneously.
__global__ __launch_bounds__(128) void gdn_decode_kernel(
    const float* __restrict__ mixed_qkv,   // (B, C)
    const float* __restrict__ a_in,        // (B, HV)
    const float* __restrict__ b_in,        // (B, HV)
    const float* __restrict__ conv_state,  // (B, 3, C)
    const float* __restrict__ ssm_state,   // (B, HV, D, D)
    const float* __restrict__ conv_w,      // (C, K)
    const float* __restrict__ dt_bias,     // (HV,)
    const float* __restrict__ A_log,       // (HV,)
    float* __restrict__ out)               // (B, HV, D)
{
    __shared__ float s_q[DD], s_k[DD], s_v[DD], s_zero[2];
    __shared__ float s_redq[128], s_redk[128];
    __shared__ float s_eg, s_beta, s_qk;

    const int bh  = blockIdx.x;
    const int b   = bh / HV;
    const int h   = bh % HV;
    const int hk  = h >> 1;        // rep = HV/HK = 2 : head h uses q/k head h/2
    const int tid = threadIdx.x;   // 0..127 == D index

    // ---- conv (K=4 taps) + SiLU for the 3 channels this thread owns ----
    const int qc = hk * DD + tid;
    const int kc = (HK + hk) * DD + tid;
    const int vc = (2 * HK + h) * DD + tid;

    auto convsilu = [&](int c) -> float {
        const float* w = conv_w + (size_t)c * KK;
        float acc = conv_state[((size_t)b * 3 + 0) * CC + c] * w[0] +
                    conv_state[((size_t)b * 3 + 1) * CC + c] * w[1] +
                    conv_state[((size_t)b * 3 + 2) * CC + c] * w[2] +
                    mixed_qkv[(size_t)b * CC + c] * w[3];
        return silu_(acc);
    };
    const float xq = convsilu(qc);
    const float xk = convsilu(kc);
    const float xv = convsilu(vc);

    // ---- L2-norm reductions for q and k (128-wide tree in LDS) ----
    s_redq[tid] = xq * xq;
    s_redk[tid] = xk * xk;
    __syncthreads();
    for (int s = 64; s > 0; s >>= 1) {
        if (tid < s) { s_redq[tid] += s_redq[tid + s]; s_redk[tid] += s_redk[tid + s]; }
        __syncthreads();
    }
    const float rq = rsqrtf(s_redq[0] + 1e-6f) * 0.08838834764831845f; // * D^-0.5
    const float rk = rsqrtf(s_redk[0] + 1e-6f);
    __syncthreads();

    const float qn = xq * rq;
    const float kn = xk * rk;
    s_q[tid] = qn;
    s_k[tid] = kn;
    s_v[tid] = xv;
    s_redq[tid] = qn * kn;            // for the k.q dot product
    if (tid < 2) s_zero[tid] = 0.0f;  // zero source for unused A-matrix rows
    if (tid == 0) {
        const float g = -__expf(A_log[h]) *
                        softplus_(a_in[(size_t)b * HV + h] + dt_bias[h]);
        s_eg   = __expf(g);
        s_beta = 1.0f / (1.0f + __expf(-b_in[(size_t)b * HV + h]));
    }
    __syncthreads();
    for (int s = 64; s > 0; s >>= 1) {
        if (tid < s) s_redq[tid] += s_redq[tid + s];
        __syncthreads();
    }
    if (tid == 0) s_qk = s_redq[0];
    __syncthreads();

    // ---- main loop: stream S through WMMA ----
    // A (16x4 f32): row 0 = q chunk, row 1 = k chunk, rows 2..15 = 0.
    //   lane layout: lane l holds A[l&15][c], A[l&15][c+1], c = (l>=16)?2:0
    // B (4x16 f32): B[kk][n] = S[v0+n][k0+kk]
    //   lane layout: lane l holds B[c][l&15], B[c+1][l&15]  -> 8B load per lane
    // D (16x16)   : D[0][n] = Sq partial, D[1][n] = Sk partial (lanes 0..15)
    const int lane  = tid & 31;
    const int wave  = tid >> 5;
    const int n     = lane & 15;
    const int chalf = (lane >> 4) << 1; // 0 or 2

    const float* abase = (n == 0) ? s_q : (n == 1) ? s_k : s_zero;
    const int    amul  = (n < 2) ? 1 : 0;

    const float* Sb = ssm_state + (size_t)bh * DD * DD;
    const float* S0 = Sb + (size_t)(wave * 32 + n) * DD + chalf;       // tile rows v0..v0+15
    const float* S1 = Sb + (size_t)(wave * 32 + 16 + n) * DD + chalf;  // tile rows v0+16..

    v8f accA = {};
    v8f accB = {};
#pragma unroll 4
    for (int k0 = 0; k0 < DD; k0 += 4) {
        v2f av;
        {
            const float* p = abase + (k0 + chalf) * amul;
            av.x = p[0];
            av.y = p[1];
        }
        const v2f sv0 = __builtin_nontemporal_load((const v2f*)(S0 + k0));
        const v2f sv1 = __builtin_nontemporal_load((const v2f*)(S1 + k0));
        accA = __builtin_amdgcn_wmma_f32_16x16x4_f32(false, av, false, sv0,
                                                     (short)0, accA, false, false);
        accB = __builtin_amdgcn_wmma_f32_16x16x4_f32(false, av, false, sv1,
                                                     (short)0, accB, false, false);
    }

    // ---- epilogue: lanes 0..15 hold (Sq, Sk) for their row in acc[0]/acc[1]
    // out[v] = eg*Sq[v] + (v[v] - eg*Sk[v]) * beta * (k.q)
    if (lane < 16) {
        const float eg = s_eg, beta = s_beta, qk = s_qk;
        const int vA = wave * 32 + lane;
        const int vB = vA + 16;
        const float oA = eg * accA[0] + (s_v[vA] - eg * accA[1]) * beta * qk;
        const float oB = eg * accB[0] + (s_v[vB] - eg * accB[1]) * beta * qk;
        float* o = out + (size_t)bh * DD;
        o[vA] = oA;
        o[vB] = oB;
    }
}

extern "C" void kernel_launch(void* const* d_in, const int* in_sizes, int n_in,
                              void* d_out, int out_size, void* d_ws, size_t ws_size,
                              hipStream_t stream) {
    (void)n_in; (void)d_ws; (void)ws_size; (void)out_size;
    const float* mixed_qkv  = (const float*)d_in[0];
    const float* a_in       = (const float*)d_in[1];
    const float* b_in       = (const float*)d_in[2];
    const float* conv_state = (const float*)d_in[3];
    const float* ssm_state  = (const float*)d_in[4];
    const float* conv_w     = (const float*)d_in[5];
    const float* dt_bias    = (const float*)d_in[6];
    const float* A_log      = (const float*)d_in[7];
    float* out = (float*)d_out;

    const int B = in_sizes[1] / HV;  // a is (B, HV)
    gdn_decode_kernel<<<dim3(B * HV), dim3(128), 0, stream>>>(
        mixed_qkv, a_in, b_in, conv_state, ssm_state, conv_w, dt_bias, A_log, out);
}